// VectorQuantizer_72164040507609
// MI455X (gfx1250) — compile-verified
//
#include <hip/hip_runtime.h>

typedef __attribute__((ext_vector_type(2))) float v2f;
typedef __attribute__((ext_vector_type(4))) float v4f;
typedef __attribute__((ext_vector_type(8))) float v8f;

#define NROWS   (64 * 64 * 64)     // 262144 flat vectors
#define CDIM    64                 // embedding dim
#define KDIM    512                // codebook size
#define PAIRSTRIDE 1056            // dwords per c-pair row: 2*512+32 (1056 % 64 == 32)
#define NPAIRS  (CDIM / 2)         // 32 c-pairs
#define Q_ELEMS (NROWS * CDIM)     // 16777216
#define WG_COUNT 2048
#define WAVES_PER_WG 8
#define TPB     (WAVES_PER_WG * 32)

// --- Kernel 1: e2[k] = sum_c E[c][k]^2 ------------------------------------
__global__ void vq_e2_kernel(const float* __restrict__ E, float* __restrict__ e2) {
  int k = blockIdx.x * blockDim.x + threadIdx.x;
  if (k < KDIM) {
    float s = 0.0f;
#pragma unroll
    for (int c = 0; c < CDIM; ++c) {
      float v = E[c * KDIM + k];
      s += v * v;
    }
    e2[k] = s;
  }
}

// --- Kernel 2: distances via V_WMMA_F32_16X16X4_F32, argmin, gather, loss --
// LDS codebook layout: pair-interleaved so each WMMA B fragment is a single
// aligned ds_load_b64 into a contiguous VGPR pair, conflict-free across all
// 64 banks (pair stride 1056 dwords, 1056 % 64 == 32).
extern __shared__ float smem[];

__global__ void __launch_bounds__(TPB, 1)
vq_main_kernel(const float* __restrict__ X, const float* __restrict__ E,
               const float* __restrict__ e2, float* __restrict__ outQ,
               float* __restrict__ outIdx, float* __restrict__ partial) {
  float* sEb   = smem;                         // NPAIRS * PAIRSTRIDE floats
  float* sPart = smem + NPAIRS * PAIRSTRIDE;   // WAVES_PER_WG floats

  const int tid = threadIdx.x;

  // Stage codebook: sEb[p*1056 + col*2 + (c&1)] = E[c][col], c = 2p, 2p+1.
  // Two b64 global loads -> one b128 LDS store (4 consecutive dest dwords).
  for (int item = tid; item < NPAIRS * (KDIM / 2); item += TPB) {
    int p  = item >> 8;            // c-pair 0..31
    int cp = item & 255;           // col-pair 0..255
    v2f r0 = *(const v2f*)(E + (2 * p)     * KDIM + 2 * cp);
    v2f r1 = *(const v2f*)(E + (2 * p + 1) * KDIM + 2 * cp);
    v4f w;
    w.x = r0.x; w.y = r1.x; w.z = r0.y; w.w = r1.y;
    *(v4f*)(sEb + p * PAIRSTRIDE + 4 * cp) = w;
  }
  __syncthreads();

  const int wave = tid >> 5;
  const int lane = tid & 31;
  const int half = lane >> 4;   // K-half for A/B fragments (ISA 16x4 layout)
  const int ln   = lane & 15;   // M (for A) / N (for B,C,D)
  const int rowBase = (blockIdx.x * WAVES_PER_WG + wave) * 16;

  // Preload A fragments: A[j] covers K = 4j..4j+3; this lane holds
  // X[rowBase+ln][4j+2*half] and X[rowBase+ln][4j+2*half+1].
  v2f A[16];
  const float* xrow = X + (size_t)(rowBase + ln) * CDIM + 2 * half;
#pragma unroll
  for (int j = 0; j < 16; ++j)
    A[j] = *(const v2f*)(xrow + 4 * j);

  // Lane-invariant part of the B-fragment LDS address (in floats):
  // pair index = 2j + half, element = (colBase + t*16 + ln)*2.
  const float* bLane = sEb + half * PAIRSTRIDE + 2 * ln;

  // Running argmin of d = ||e||^2 - 2 * x.e  (||x||^2 constant per row).
  float minV[8];
  int   minI[8];
#pragma unroll
  for (int v = 0; v < 8; ++v) { minV[v] = 3.4e38f; minI[v] = 0; }

  for (int g = 0; g < 8; ++g) {           // 8 groups x 4 tiles x 16 cols = 512
    v8f acc[4];
    const v8f zero = {};
#pragma unroll
    for (int t = 0; t < 4; ++t) acc[t] = zero;
    const int colBase = g * 64;
#pragma unroll
    for (int j = 0; j < 16; ++j) {
      const float* bj = bLane + j * (2 * PAIRSTRIDE) + colBase * 2;
#pragma unroll
      for (int t = 0; t < 4; ++t) {
        v2f b = *(const v2f*)(bj + t * 32);  // {E[4j+2h][col], E[4j+2h+1][col]}
        acc[t] = __builtin_amdgcn_wmma_f32_16x16x4_f32(
            false, A[j], false, b, (short)0, acc[t], false, false);
      }
    }
#pragma unroll
    for (int t = 0; t < 4; ++t) {
      const int col = colBase + t * 16 + ln;
      const float ev = e2[col];
#pragma unroll
      for (int v = 0; v < 8; ++v) {
        float d = ev - 2.0f * acc[t][v];  // lane's col, row = v + 8*half
        if (d < minV[v]) { minV[v] = d; minI[v] = col; }  // cols increase -> first-min kept
      }
    }
  }

  // Butterfly argmin across the 16 lanes of this half (xor<16 stays in-half).
#pragma unroll
  for (int off = 8; off >= 1; off >>= 1) {
#pragma unroll
    for (int v = 0; v < 8; ++v) {
      float ov = __shfl_xor(minV[v], off, 32);
      int   oi = __shfl_xor(minI[v], off, 32);
      if (ov < minV[v] || (ov == minV[v] && oi < minI[v])) {
        minV[v] = ov; minI[v] = oi;
      }
    }
  }

  // Write encoding indices (row = slot + 8*half); output buffer is float.
  if (ln < 8) {
    int row = half * 8 + ln;
    outIdx[rowBase + row] = (float)minI[ln];
  }

  // Gather quantized rows from the interleaved LDS codebook, write, loss.
  float lsum = 0.0f;
  for (int r = 0; r < 16; ++r) {
    const int src = ((r >> 3) << 4) + (r & 7);
    const int idx = __shfl(minI[r & 7], src, 32);
    const float* xr = X    + (size_t)(rowBase + r) * CDIM;
    float*       qr = outQ + (size_t)(rowBase + r) * CDIM;
#pragma unroll
    for (int c = lane; c < CDIM; c += 32) {
      float q = sEb[(c >> 1) * PAIRSTRIDE + 2 * idx + (c & 1)];
      float x = xr[c];
      qr[c] = q;                 // quantized_st == quantized numerically
      float df = x - q;
      lsum += df * df;
    }
  }

  // Deterministic block partial sum.
#pragma unroll
  for (int off = 16; off >= 1; off >>= 1)
    lsum += __shfl_xor(lsum, off, 32);
  if (lane == 0) sPart[wave] = lsum;
  __syncthreads();
  if (tid == 0) {
    float s = 0.0f;
#pragma unroll
    for (int w = 0; w < WAVES_PER_WG; ++w) s += sPart[w];
    partial[blockIdx.x] = s;
  }
}

// --- Kernel 3: fixed-order final reduction -> two scalar losses ------------
__global__ void vq_loss_kernel(const float* __restrict__ partial,
                               float* __restrict__ outLoss) {
  __shared__ float red[256];
  float s = 0.0f;
  for (int i = threadIdx.x; i < WG_COUNT; i += 256) s += partial[i];
  red[threadIdx.x] = s;
  __syncthreads();
  for (int stride = 128; stride >= 1; stride >>= 1) {
    if ((int)threadIdx.x < stride) red[threadIdx.x] += red[threadIdx.x + stride];
    __syncthreads();
  }
  if (threadIdx.x == 0) {
    float mean = red[0] / (float)Q_ELEMS;
    outLoss[0] = 0.25f * mean;  // commitment_loss
    outLoss[1] = mean;          // codebook_loss (same value, unscaled)
  }
}

extern "C" void kernel_launch(void* const* d_in, const int* in_sizes, int n_in,
                              void* d_out, int out_size, void* d_ws, size_t ws_size,
                              hipStream_t stream) {
  const float* X = (const float*)d_in[0];  // (64,64,64,64) f32
  const float* E = (const float*)d_in[1];  // (64,512) f32

  float* outQ    = (float*)d_out;          // 16777216 quantized_st
  float* outIdx  = outQ + Q_ELEMS;         // 262144 indices (as float)
  float* outLoss = outIdx + NROWS;         // 2 scalars

  float* e2      = (float*)d_ws;           // 512 floats
  float* partial = e2 + KDIM;              // 2048 floats

  const size_t shbytes =
      (size_t)(NPAIRS * PAIRSTRIDE + WAVES_PER_WG) * sizeof(float);  // 135200 B
  // Allow >64KB dynamic LDS (CDNA5 WGP has 320KB).
  (void)hipFuncSetAttribute((const void*)vq_main_kernel,
                            hipFuncAttributeMaxDynamicSharedMemorySize,
                            (int)shbytes);

  vq_e2_kernel<<<(KDIM + 255) / 256, 256, 0, stream>>>(E, e2);
  vq_main_kernel<<<WG_COUNT, TPB, shbytes, stream>>>(X, E, e2, outQ, outIdx, partial);
  vq_loss_kernel<<<1, 256, 0, stream>>>(partial, outLoss);
}